// _FP8W8A8Linear_52372831207457
// MI455X (gfx1250) — compile-verified
//
#include <hip/hip_runtime.h>

typedef float v2f __attribute__((ext_vector_type(2)));
typedef float v8f __attribute__((ext_vector_type(8)));
typedef int   v4i __attribute__((ext_vector_type(4)));

#define D_K   4096      // inner dim
#define D_O   4096      // output features
#define NBLK  (D_K / 128)
#define TM    128       // workgroup M tile
#define TN    128       // workgroup N tile
#define BK    16        // K tile staged in LDS
#define KSTR  20        // padded LDS row stride (floats): 20*L mod 64 distinct for L=0..15
#define NSTG  3         // LDS pipeline stages (2 tiles in flight)

#if __has_builtin(__builtin_amdgcn_global_load_async_to_lds_b128)
#define USE_ASYNC 1
#else
#define USE_ASYNC 0
#endif

#if USE_ASYNC
  #if __has_builtin(__builtin_amdgcn_s_wait_asynccnt)
    #define WAIT_ASYNC(n) __builtin_amdgcn_s_wait_asynccnt(n)
  #else
    #define WAIT_ASYNC(n) asm volatile("s_wait_asynccnt %0" :: "i"(n) : "memory")
  #endif
#else
  #define WAIT_ASYNC(n) ((void)0)
#endif

__device__ __forceinline__ float bf16_rne(float x) {
    unsigned u = __float_as_uint(x);
    u = (u + 0x7fffu + ((u >> 16) & 1u)) & 0xffff0000u;
    return __uint_as_float(u);
}

// 16-byte global -> LDS copy: async (ASYNCcnt) on CDNA5, sync fallback otherwise.
__device__ __forceinline__ void cp16(const float* g, float* l) {
#if USE_ASYNC
    __builtin_amdgcn_global_load_async_to_lds_b128(
        (__attribute__((address_space(1))) v4i*)(v4i*)g,
        (__attribute__((address_space(3))) v4i*)(v4i*)l,
        0 /*imm offset*/, 0 /*cpol*/);
#else
    *(float4*)l = *(const float4*)g;
#endif
}

__global__ __launch_bounds__(256) void _FP8W8A8Linear_52372831207457_kernel(
    const float* __restrict__ X,       // [T, 4096]
    const float* __restrict__ W,       // [4096, 4096] fp8 codes (f32)
    const float* __restrict__ Wsc,     // [4096, 32]
    const float* __restrict__ bias,    // [4096]
    float* __restrict__ Y)             // [T, 4096]
{
    __shared__ float Xs[NSTG][TM * KSTR];   // 3 x 10 KB
    __shared__ float Wl[NSTG][TM * KSTR];   // 3 x 10 KB (raw fp8 codes as f32)

    const int tid   = threadIdx.x;
    const int lane  = tid & 31;             // wave32
    const int wave  = tid >> 5;             // 0..7
    const int waveM = wave & 3;             // 32-row stripe
    const int waveN = wave >> 2;            // 64-col stripe

    const int m0 = blockIdx.y * TM;
    const int n0 = blockIdx.x * TN;

    const int l16  = lane & 15;
    const int half = lane >> 4;             // K pair {k,k+1} vs {k+2,k+3}

    // ---- per-tile copy: 128 rows x 16 floats = 512 float4; 2 per thread per matrix ----
    auto issue_tile = [&](int stage, int k0) {
        #pragma unroll
        for (int j = 0; j < 2; ++j) {
            const int f   = tid + 256 * j;
            const int row = f >> 2;
            const int c4  = (f & 3) * 4;
            cp16(X + (size_t)(m0 + row) * D_K + k0 + c4, &Xs[stage][row * KSTR + c4]);
            cp16(W + (size_t)(n0 + row) * D_K + k0 + c4, &Wl[stage][row * KSTR + c4]);
        }
    };

    v8f acc[2][4] = {};                     // 2(M) x 4(N) 16x16 f32 accumulators
    float sc[4];                            // per-lane bf16-rounded W scales (per N tile)

    const int NT = D_K / BK;                // 256 K-tiles

    issue_tile(0, 0);
    issue_tile(1, BK);

    for (int t = 0; t < NT; ++t) {
        // refresh W scales on each new 128-wide K block (every 8 tiles)
        if ((t & 7) == 0) {
            const int kb = (t * BK) >> 7;
            #pragma unroll
            for (int ni = 0; ni < 4; ++ni) {
                const int row = n0 + waveN * 64 + ni * 16 + l16;
                sc[ni] = bf16_rne(Wsc[(size_t)row * NBLK + kb]);
            }
        }

        // retire the oldest in-flight tile (4 wave-wide async instrs per tile,
        // async loads complete in order), then make it visible to all waves
        if (t + 1 < NT) { WAIT_ASYNC(4); } else { WAIT_ASYNC(0); }
        __syncthreads();

        if (t + 2 < NT) issue_tile((t + 2) % NSTG, (t + 2) * BK);

        const float* xs = Xs[t % NSTG];
        const float* ws = Wl[t % NSTG];

        #pragma unroll
        for (int ks = 0; ks < BK; ks += 4) {
            const int koff = ks + 2 * half;
            v2f a[2], b[4];
            #pragma unroll
            for (int mi = 0; mi < 2; ++mi) {
                const int row = waveM * 32 + mi * 16 + l16;
                a[mi] = *(const v2f*)(&xs[row * KSTR + koff]);
            }
            #pragma unroll
            for (int ni = 0; ni < 4; ++ni) {
                const int row = waveN * 64 + ni * 16 + l16;
                b[ni] = *(const v2f*)(&ws[row * KSTR + koff]);
                b[ni] = b[ni] * sc[ni];     // fused weight dequant (co-executes with WMMA)
            }
            #pragma unroll
            for (int mi = 0; mi < 2; ++mi)
                #pragma unroll
                for (int ni = 0; ni < 4; ++ni)
                    acc[mi][ni] = __builtin_amdgcn_wmma_f32_16x16x4_f32(
                        false, a[mi], false, b[ni],
                        (short)0, acc[mi][ni], false, false);
        }
        __syncthreads();
    }

    // ---- epilogue: bias add + store (C/D layout: row = v + 8*half, col = l16) ----
    #pragma unroll
    for (int ni = 0; ni < 4; ++ni) {
        const int col = n0 + waveN * 64 + ni * 16 + l16;
        const float bv = bias[col];
        #pragma unroll
        for (int mi = 0; mi < 2; ++mi) {
            const int rbase = m0 + waveM * 32 + mi * 16 + 8 * half;
            #pragma unroll
            for (int v = 0; v < 8; ++v) {
                Y[(size_t)(rbase + v) * D_O + col] = acc[mi][ni][v] + bv;
            }
        }
    }
}

extern "C" void kernel_launch(void* const* d_in, const int* in_sizes, int n_in,
                              void* d_out, int out_size, void* d_ws, size_t ws_size,
                              hipStream_t stream) {
    const float* X    = (const float*)d_in[0];
    const float* W    = (const float*)d_in[1];
    const float* Wsc  = (const float*)d_in[2];
    const float* bias = (const float*)d_in[3];
    float* Y = (float*)d_out;

    const int T = in_sizes[0] / D_K;            // 8192 tokens
    dim3 grid(D_O / TN, T / TM);                // (32, 64)
    _FP8W8A8Linear_52372831207457_kernel<<<grid, 256, 0, stream>>>(X, W, Wsc, bias, Y);
}